// AGGLayer_73976516706889
// MI455X (gfx1250) — compile-verified
//
#include <hip/hip_runtime.h>
#include <cstdint>

#define WAVES_PER_BLOCK 8
#define PIPE 4u
#define ASYNC_WAIT(n) asm volatile("s_wait_asynccnt %0" :: "n"(n) : "memory")

// ---------------- Phase 0: zero degree counters ----------------
__global__ void k_zero_u32(unsigned* __restrict__ p, int n) {
  int i = blockIdx.x * blockDim.x + threadIdx.x;
  if (i < n) p[i] = 0u;
}

// ---------------- Phase 1: degree histogram (E u32 atomics) ----------------
__global__ void k_degree(const long long* __restrict__ dst,
                         unsigned* __restrict__ deg, int E) {
  int i = blockIdx.x * blockDim.x + threadIdx.x;
  if (i < E) atomicAdd(&deg[(unsigned)dst[i]], 1u);
}

// ---------------- Phase 2a: per-block exclusive scan (wave32 shuffles) -----
// 256 threads = 8 waves; 2 barriers per block total.
__global__ __launch_bounds__(256) void k_scan_block(
    const unsigned* __restrict__ deg, unsigned* __restrict__ offs,
    unsigned* __restrict__ bsums, int N) {
  __shared__ unsigned wsum[8];
  const int t = threadIdx.x;
  const int lane = t & 31, w = t >> 5;
  const int i = blockIdx.x * 256 + t;
  unsigned v = (i < N) ? deg[i] : 0u;

  // inclusive scan within wave
  unsigned x = v;
  #pragma unroll
  for (int d = 1; d < 32; d <<= 1) {
    unsigned y = __shfl_up(x, d, 32);
    if (lane >= d) x += y;
  }
  if (lane == 31) wsum[w] = x;
  __syncthreads();
  if (w == 0) {
    unsigned s = (lane < 8) ? wsum[lane] : 0u;
    #pragma unroll
    for (int d = 1; d < 8; d <<= 1) {
      unsigned y = __shfl_up(s, d, 32);
      if (lane >= d) s += y;
    }
    if (lane < 8) wsum[lane] = s;  // inclusive scan of wave sums
  }
  __syncthreads();
  unsigned waveBase = (w > 0) ? wsum[w - 1] : 0u;
  if (i < N) offs[i] = waveBase + x - v;            // exclusive within block
  if (t == 255) bsums[blockIdx.x] = waveBase + x;    // block total
}

// ---------------- Phase 2b: single-block scan of block totals --------------
__global__ __launch_bounds__(1024) void k_scan_top(unsigned* __restrict__ a, int B) {
  __shared__ unsigned sm[1024];
  __shared__ unsigned carry_s;
  const int t = threadIdx.x;
  if (t == 0) carry_s = 0u;
  __syncthreads();
  for (int base = 0; base < B; base += 1024) {
    int i = base + t;
    unsigned v = (i < B) ? a[i] : 0u;
    sm[t] = v;
    __syncthreads();
    for (int off = 1; off < 1024; off <<= 1) {
      unsigned add = (t >= off) ? sm[t - off] : 0u;
      __syncthreads();
      sm[t] += add;
      __syncthreads();
    }
    unsigned incl = sm[t];
    unsigned carry = carry_s;
    if (i < B) a[i] = carry + incl - v;   // exclusive, in place
    __syncthreads();
    if (t == 1023) carry_s = carry + incl;
    __syncthreads();
  }
}

// ---------------- Phase 2c: add block bases; init cursors ------------------
__global__ void k_scan_add(unsigned* __restrict__ offs, unsigned* __restrict__ cursor,
                           const unsigned* __restrict__ bsums, int N) {
  int i = blockIdx.x * blockDim.x + threadIdx.x;
  if (i < N) {
    unsigned x = offs[i] + bsums[i >> 8];
    offs[i] = x;
    cursor[i] = x;
  }
}

// ---------------- Phase 3: bucket-scatter edge ids ----------------
__global__ void k_scatter(const long long* __restrict__ dst,
                          unsigned* __restrict__ cursor,
                          unsigned* __restrict__ elist, int E) {
  int i = blockIdx.x * blockDim.x + threadIdx.x;
  if (i < E) {
    unsigned d = (unsigned)dst[i];
    unsigned p = atomicAdd(&cursor[d], 1u);
    elist[p] = (unsigned)i;
  }
}

// ---------------- Phase 4: per-node gather + mean ----------------
// One wave per node; 32 lanes x float2 = 64 features. Edge rows stream through
// a 4-deep ring of LDS buffers via gfx1250 async global->LDS DMA (2 rows =
// 512B per b128 async op per wave), drained with s_wait_asynccnt. src rows are
// gathered directly (src_embedding is L2-resident) and prefetched at issue.
__global__ __launch_bounds__(256) void k_gather_mean(
    const float* __restrict__ src_emb, const float* __restrict__ edge_emb,
    const long long* __restrict__ src_idx,
    const unsigned* __restrict__ offs, const unsigned* __restrict__ deg,
    const unsigned* __restrict__ elist, float* __restrict__ out, int N) {
  // per wave: 4 buffers * 2 rows * 64 floats = 512 floats (2 KB)
  __shared__ __align__(16) float stage[WAVES_PER_BLOCK * 512];
  const int wib = threadIdx.x >> 5;
  const int lane = threadIdx.x & 31;
  const int node = blockIdx.x * WAVES_PER_BLOCK + wib;
  if (node >= N) return;

  float* wstage = &stage[wib * 512];
  const unsigned ldsBase =
      (unsigned)(unsigned long long)(__attribute__((address_space(3))) float*)wstage;

  unsigned start = __builtin_amdgcn_readfirstlane(offs[node]);
  unsigned cnt   = __builtin_amdgcn_readfirstlane(deg[node]);
  const unsigned npairs = (cnt + 1u) >> 1;

  const unsigned rowSel = (unsigned)(lane >> 4);  // 0 -> first row, 1 -> second
  const unsigned quad   = (unsigned)(lane & 15);  // 16B chunk within row

  float2 acc; acc.x = 0.f; acc.y = 0.f;

  auto issue = [&](unsigned q) {
    unsigned j0 = start + 2u * q;
    unsigned e0 = elist[j0];
    unsigned e1 = (j0 + 1u < start + cnt) ? elist[j0 + 1u] : e0;
    unsigned row = rowSel ? e1 : e0;
    const float* g = edge_emb + (size_t)row * 64 + quad * 4;
    unsigned lds = ldsBase + (q & 3u) * 512u + (unsigned)lane * 16u;
    asm volatile("global_load_async_to_lds_b128 %0, %1, off"
                 :: "v"(lds), "v"((unsigned long long)g) : "memory");
    // warm caches for the dependent src gathers of this pair
    long long s0 = src_idx[e0];
    __builtin_prefetch(src_emb + (size_t)s0 * 64 + lane * 2, 0, 3);
    long long s1 = src_idx[e1];
    __builtin_prefetch(src_emb + (size_t)s1 * 64 + lane * 2, 0, 3);
  };

  const unsigned pre = (npairs < PIPE) ? npairs : PIPE;
  for (unsigned q = 0; q < pre; ++q) issue(q);

  for (unsigned p = 0; p < npairs; ++p) {
    // outstanding after wait must be <= (#pairs issued) - (p+1)
    unsigned remaining = npairs - 1u - p;
    unsigned wv = (remaining < (PIPE - 1u)) ? remaining : (PIPE - 1u);
    switch (wv) {                       // wave-uniform
      case 0u: ASYNC_WAIT(0); break;
      case 1u: ASYNC_WAIT(1); break;
      case 2u: ASYNC_WAIT(2); break;
      default: ASYNC_WAIT(3); break;
    }

    const float* buf = wstage + (p & 3u) * 128u;
    unsigned j0 = start + 2u * p;
    unsigned e0 = elist[j0];
    {
      long long s = src_idx[e0];
      const float2 a = *(const float2*)(src_emb + (size_t)s * 64 + lane * 2);
      const float2 m = *(const float2*)(buf + lane * 2);
      acc.x += a.x + m.x;
      acc.y += a.y + m.y;
    }
    if (2u * p + 1u < cnt) {
      unsigned e1 = elist[j0 + 1u];
      long long s = src_idx[e1];
      const float2 a = *(const float2*)(src_emb + (size_t)s * 64 + lane * 2);
      const float2 m = *(const float2*)(buf + 64 + lane * 2);
      acc.x += a.x + m.x;
      acc.y += a.y + m.y;
    }
    if (p + PIPE < npairs) issue(p + PIPE);  // ring slot (p&3) is free now
  }

  const float inv = 1.0f / (float)((cnt > 0u) ? cnt : 1u);
  float2 r; r.x = acc.x * inv; r.y = acc.y * inv;
  *(float2*)(out + (size_t)node * 64 + lane * 2) = r;
}

extern "C" void kernel_launch(void* const* d_in, const int* in_sizes, int n_in,
                              void* d_out, int out_size, void* d_ws, size_t ws_size,
                              hipStream_t stream) {
  (void)n_in; (void)out_size; (void)ws_size;
  const float*     src_emb  = (const float*)d_in[0];
  const float*     edge_emb = (const float*)d_in[1];
  const long long* src_idx  = (const long long*)d_in[2];
  const long long* dst_idx  = (const long long*)d_in[3];

  const int E = in_sizes[2];
  const int D = in_sizes[1] / E;   // 64
  const int N = in_sizes[0] / D;   // 100000
  const int B = (N + 255) / 256;   // number of scan blocks
  float* out = (float*)d_out;

  // workspace: deg[N], offs[N], cursor[N], elist[E], bsums[B]  (~6.2 MB)
  unsigned* deg    = (unsigned*)d_ws;
  unsigned* offs   = deg + N;
  unsigned* cursor = offs + N;
  unsigned* elist  = cursor + N;
  unsigned* bsums  = elist + E;

  const int tb = 256;
  k_zero_u32  <<<(N + tb - 1) / tb, tb, 0, stream>>>(deg, N);
  k_degree    <<<(E + tb - 1) / tb, tb, 0, stream>>>(dst_idx, deg, E);
  k_scan_block<<<B, 256, 0, stream>>>(deg, offs, bsums, N);
  k_scan_top  <<<1, 1024, 0, stream>>>(bsums, B);
  k_scan_add  <<<(N + tb - 1) / tb, tb, 0, stream>>>(offs, cursor, bsums, N);
  k_scatter   <<<(E + tb - 1) / tb, tb, 0, stream>>>(dst_idx, cursor, elist, E);
  k_gather_mean<<<(N + WAVES_PER_BLOCK - 1) / WAVES_PER_BLOCK,
                  32 * WAVES_PER_BLOCK, 0, stream>>>(
      src_emb, edge_emb, src_idx, offs, deg, elist, out, N);
}